// GraphNN_9775345566165
// MI455X (gfx1250) — compile-verified
//
#include <hip/hip_runtime.h>
#include <cmath>

#define N_NODES 50000
#define N_EDGES 600000
#define D 128
#define ITERS 3

typedef __attribute__((ext_vector_type(2))) float v2f;
typedef __attribute__((ext_vector_type(8))) float v8f;

// ---------------------------------------------------------------- degree ----
__global__ void init_denom_kernel(float* __restrict__ denom) {
    int i = blockIdx.x * blockDim.x + threadIdx.x;
    if (i < N_NODES) denom[i] = 1.0f;   // deg + 1
}

__global__ void degree_kernel(const long long* __restrict__ adj,
                              float* __restrict__ denom) {
    int e = blockIdx.x * blockDim.x + threadIdx.x;
    if (e < N_EDGES) {
        int s = (int)adj[2 * (size_t)e];
        int d = (int)adj[2 * (size_t)e + 1];
        atomicAdd(&denom[s], 1.0f);
        atomicAdd(&denom[d], 1.0f);
    }
}

// ---------------------------------------------------------------- scatter ---
// One wave32 per edge: lanes cover 128 dims as 4 coalesced strides of 32.
__global__ void scatter_kernel(const long long* __restrict__ adj,
                               const float* __restrict__ h,
                               float* __restrict__ agg) {
    int gw   = (int)(((size_t)blockIdx.x * blockDim.x + threadIdx.x) >> 5);
    int lane = threadIdx.x & 31;
    if (gw >= N_EDGES) return;
    int s = (int)adj[2 * (size_t)gw];
    int d = (int)adj[2 * (size_t)gw + 1];
    const float* hs = h + (size_t)s * D;
    const float* hd = h + (size_t)d * D;
    float* as = agg + (size_t)s * D;
    float* ad = agg + (size_t)d * D;
#pragma unroll
    for (int j = 0; j < 4; ++j) {
        int c = lane + 32 * j;
        atomicAdd(&as[c], hd[c]);   // agg[src] += h[dst]
        atomicAdd(&ad[c], hs[c]);   // agg[dst] += h[src]
    }
}

// ---------------------------------------------------------------- GEMM ------
// out[M x 128] = act( pre(Ain) @ W + bias ), tile: block = 16 rows, wave = 16 cols.
// ACT: 0 = relu, 1 = sigmoid.  FUSE: pre = (Ain + agg) / denom (row-wise).
// Safe in-place (out == Ain): A-tile fully staged to LDS before any store.
template <int ACT, bool FUSE>
__global__ __launch_bounds__(256) void gemm_wmma_kernel(
    const float* __restrict__ Ain, const float* __restrict__ W,
    const float* __restrict__ bias, float* __restrict__ out,
    const float* __restrict__ agg, const float* __restrict__ denom) {
    // pitch 132: 132 % 64 == 4 -> the 16 lanes of an A-fragment hit distinct banks
    __shared__ float As[16 * 132];

    const int row0 = blockIdx.x * 16;
    const int tid  = threadIdx.x;
    const int wave = tid >> 5;      // 0..7 -> column tile
    const int lane = tid & 31;

    // Stage 16 x 128 A tile (fused normalization for the update GEMM).
    for (int i = tid; i < 16 * D; i += 256) {
        int r = i >> 7, c = i & (D - 1);
        size_t gi = (size_t)(row0 + r) * D + c;
        float v = Ain[gi];
        if constexpr (FUSE) v = (v + agg[gi]) / denom[row0 + r];
        As[r * 132 + c] = v;
    }
    __syncthreads();

    const int n0   = wave << 4;
    const int half = lane >> 4;     // 0: K pair {0,1}; 1: K pair {2,3}
    const int lr   = lane & 15;

    v8f acc = {};
#pragma unroll
    for (int k = 0; k < D; k += 4) {
        const int ka = k + 2 * half;
        v2f a, b;
        // A frag: lane l holds A[l%16][ka], A[l%16][ka+1]
        a.x = As[lr * 132 + ka];
        a.y = As[lr * 132 + ka + 1];
        // B frag: lane l holds W[ka][n0+l%16], W[ka+1][n0+l%16]
        b.x = W[(size_t)ka * D + n0 + lr];
        b.y = W[(size_t)(ka + 1) * D + n0 + lr];
        acc = __builtin_amdgcn_wmma_f32_16x16x4_f32(
            /*neg_a=*/false, a, /*neg_b=*/false, b,
            /*c_mod=*/(short)0, acc, /*reuse_a=*/false, /*reuse_b=*/false);
    }

    // D layout: lane l, vgpr v -> D[v + 8*(l/16)][n0 + l%16]
    const float bn = bias[n0 + lr];
#pragma unroll
    for (int v = 0; v < 8; ++v) {
        float x = acc[v] + bn;
        if (ACT == 0)
            x = fmaxf(x, 0.0f);
        else
            x = 1.0f / (1.0f + expf(-x));
        out[(size_t)(row0 + v + 8 * half) * D + n0 + lr] = x;
    }
}

// ---------------------------------------------------------------- launch ----
extern "C" void kernel_launch(void* const* d_in, const int* in_sizes, int n_in,
                              void* d_out, int out_size, void* d_ws, size_t ws_size,
                              hipStream_t stream) {
    const float*     node_feats = (const float*)d_in[0];
    const float*     W_emb      = (const float*)d_in[1];
    const float*     b_emb      = (const float*)d_in[2];
    const float*     W_upd      = (const float*)d_in[3];
    const float*     b_upd      = (const float*)d_in[4];
    const long long* adj        = (const long long*)d_in[5];

    float* h     = (float*)d_out;                                   // 25.6 MB
    float* agg   = (float*)d_ws;                                    // 25.6 MB
    float* denom = (float*)((char*)d_ws + (size_t)N_NODES * D * 4); // 200 KB

    init_denom_kernel<<<(N_NODES + 255) / 256, 256, 0, stream>>>(denom);
    degree_kernel<<<(N_EDGES + 255) / 256, 256, 0, stream>>>(adj, denom);

    // h = relu(node_feats @ W_emb + b_emb)
    gemm_wmma_kernel<0, false><<<N_NODES / 16, 256, 0, stream>>>(
        node_feats, W_emb, b_emb, h, nullptr, nullptr);

    for (int it = 0; it < ITERS; ++it) {
        hipMemsetAsync(agg, 0, (size_t)N_NODES * D * sizeof(float), stream);
        scatter_kernel<<<(N_EDGES + 7) / 8, 256, 0, stream>>>(adj, h, agg);
        // h = sigmoid(((h + agg) / denom) @ W_upd + b_upd), in place
        gemm_wmma_kernel<1, true><<<N_NODES / 16, 256, 0, stream>>>(
            h, W_upd, b_upd, h, agg, denom);
    }
}